// MSSSIM_180388626877
// MI455X (gfx1250) — compile-verified
//
#include <hip/hip_runtime.h>

typedef __attribute__((ext_vector_type(2))) float v2f;
typedef __attribute__((ext_vector_type(8))) float v8f;

#define TILE      32
#define HALO      10
#define IN_DIM    42          // TILE + HALO
#define IN_STRIDE 44
#define HB_STRIDE 48
#define NMAPS     5
#define KCHUNKS   7           // ceil(26/4) K-chunks of 4 for the banded matmul
#define C1F       (1.0e-4f)   // 0.01^2
#define C2F       (9.0e-4f)   // 0.03^2

// Normalized 11-tap Gaussian, sigma = 1.5
__constant__ float GW[11] = {
    0.00102838f, 0.00759893f, 0.03600077f, 0.10936213f, 0.21300553f,
    0.26601188f,
    0.21300553f, 0.10936213f, 0.03600077f, 0.00759893f, 0.00102838f
};

__global__ void msssim_zero_accum(float* a) {
    if (threadIdx.x < 16) a[threadIdx.x] = 0.0f;
}

// One workgroup = one 32x32 output tile of one (batch,channel) plane.
// Fused: load halo -> 2x2 avg-pool emit -> horizontal Gaussian (5 maps, VALU)
//        -> vertical Gaussian as banded GEMM via V_WMMA_F32_16X16X4_F32
//        -> SSIM/CS pointwise + masked reduction -> 2 float atomics.
__global__ __launch_bounds__(128) void msssim_level(
    const float* __restrict__ img1, const float* __restrict__ img2,
    float* __restrict__ pool1, float* __restrict__ pool2,
    float* __restrict__ accum2, int H, int W)
{
    __shared__ float sIn1[IN_DIM][IN_STRIDE];
    __shared__ float sIn2[IN_DIM][IN_STRIDE];
    __shared__ float sHb[NMAPS][IN_DIM][HB_STRIDE];
    __shared__ float sRed[8];

    const int tid   = threadIdx.x;
    const int x0    = blockIdx.x * TILE;
    const int y0    = blockIdx.y * TILE;
    const int plane = blockIdx.z;
    const size_t pbase = (size_t)plane * H * W;

    // ---- load 42x42 halo tiles of both images (zero-fill OOB) ----
    for (int i = tid; i < IN_DIM * IN_DIM; i += 128) {
        int r = i / IN_DIM, c = i - r * IN_DIM;
        int gy = y0 + r, gx = x0 + c;
        bool inb = (gy < H) && (gx < W);
        size_t idx = pbase + (size_t)gy * W + gx;
        sIn1[r][c] = inb ? img1[idx] : 0.0f;
        sIn2[r][c] = inb ? img2[idx] : 0.0f;
    }
    __syncthreads();

    // ---- fused 2x2 average pool of the interior (next pyramid level) ----
    if (pool1 != nullptr) {
        int Hp = H >> 1, Wp = W >> 1;
        size_t ppbase = (size_t)plane * Hp * Wp;
        for (int i = tid; i < 16 * 16; i += 128) {
            int pr = i >> 4, pc = i & 15;
            int r = pr * 2, c = pc * 2;
            float a = 0.25f * (sIn1[r][c] + sIn1[r][c+1] + sIn1[r+1][c] + sIn1[r+1][c+1]);
            float b = 0.25f * (sIn2[r][c] + sIn2[r][c+1] + sIn2[r+1][c] + sIn2[r+1][c+1]);
            size_t o = ppbase + (size_t)((y0 >> 1) + pr) * Wp + ((x0 >> 1) + pc);
            pool1[o] = a;
            pool2[o] = b;
        }
    }

    // ---- horizontal Gaussian pass: 5 maps at 42 rows x 32 cols ----
    for (int i = tid; i < IN_DIM * TILE; i += 128) {
        int r = i >> 5, c = i & 31;
        float s1 = 0.f, s2 = 0.f, s11 = 0.f, s22 = 0.f, s12 = 0.f;
        #pragma unroll
        for (int j = 0; j < 11; ++j) {
            float g = GW[j];
            float a = sIn1[r][c + j], b = sIn2[r][c + j];
            s1  += g * a;      s2  += g * b;
            s11 += g * a * a;  s22 += g * b * b;  s12 += g * a * b;
        }
        sHb[0][r][c] = s1;  sHb[1][r][c] = s2;
        sHb[2][r][c] = s11; sHb[3][r][c] = s22; sHb[4][r][c] = s12;
    }
    __syncthreads();

    // ---- vertical Gaussian pass as banded GEMM (WMMA f32 16x16x4) ----
    // Each wave owns one 16x16 output subtile; O = A(16x28 band) x B(28x16 rows).
    const int wave = tid >> 5;
    const int lane = tid & 31;
    const int sr   = (wave >> 1) * 16;       // subtile row offset in tile
    const int sc   = (wave & 1) * 16;        // subtile col offset in tile
    const int mrow = lane & 15;              // A-matrix M per ISA 16x4 layout
    const int kb   = (lane >> 4) << 1;       // K base: 0 (lanes 0-15) / 2 (16-31)
    const int bcol = sc + (lane & 15);       // B/D column per lane

    // A fragments: banded Gaussian A[m][k] = g[k-m], identical for all maps.
    v2f afrag[KCHUNKS];
    #pragma unroll
    for (int kk = 0; kk < KCHUNKS; ++kk) {
        int k0 = kk * 4 + kb;
        int d0 = k0 - mrow, d1 = k0 + 1 - mrow;
        afrag[kk].x = (d0 >= 0 && d0 <= 10) ? GW[d0] : 0.0f;
        afrag[kk].y = (d1 >= 0 && d1 <= 10) ? GW[d1] : 0.0f;
    }

    v8f acc[NMAPS];
    #pragma unroll
    for (int m = 0; m < NMAPS; ++m) {
        v8f c = {};
        #pragma unroll
        for (int kk = 0; kk < KCHUNKS; ++kk) {
            int krow = sr + kk * 4 + kb;                 // B row K per lane group
            int kr0 = (krow     < IN_DIM) ? krow     : (IN_DIM - 1); // pad K=26,27
            int kr1 = (krow + 1 < IN_DIM) ? krow + 1 : (IN_DIM - 1); // (weights=0)
            v2f b;
            b.x = sHb[m][kr0][bcol];
            b.y = sHb[m][kr1][bcol];
            c = __builtin_amdgcn_wmma_f32_16x16x4_f32(
                    /*neg_a=*/false, afrag[kk], /*neg_b=*/false, b,
                    /*c_mod=*/(short)0, c, /*reuse_a=*/false, /*reuse_b=*/false);
        }
        acc[m] = c;
    }

    // ---- pointwise SSIM/CS + masked mean contribution ----
    float sum_ssim = 0.0f, sum_cs = 0.0f;
    const int Hv = H - HALO, Wv = W - HALO;   // valid conv output extent
    #pragma unroll
    for (int v = 0; v < 8; ++v) {
        int r  = sr + v + ((lane >> 4) << 3); // D layout: M = v (+8 for hi lanes)
        int gy = y0 + r, gx = x0 + bcol;
        if (gy < Hv && gx < Wv) {
            float mu1 = acc[0][v], mu2 = acc[1][v];
            float e11 = acc[2][v], e22 = acc[3][v], e12 = acc[4][v];
            float mu1s = mu1 * mu1, mu2s = mu2 * mu2, mu12 = mu1 * mu2;
            float s1 = e11 - mu1s, s2 = e22 - mu2s, s12 = e12 - mu12;
            float v1 = 2.0f * s12 + C2F;
            float v2 = s1 + s2 + C2F;
            sum_cs   += v1 / v2;
            sum_ssim += ((2.0f * mu12 + C1F) * v1) / ((mu1s + mu2s + C1F) * v2);
        }
    }
    #pragma unroll
    for (int off = 16; off > 0; off >>= 1) {
        sum_ssim += __shfl_down(sum_ssim, off, 32);
        sum_cs   += __shfl_down(sum_cs,   off, 32);
    }
    if (lane == 0) { sRed[wave] = sum_ssim; sRed[4 + wave] = sum_cs; }
    __syncthreads();
    if (tid == 0) {
        atomicAdd(&accum2[0], sRed[0] + sRed[1] + sRed[2] + sRed[3]);
        atomicAdd(&accum2[1], sRed[4] + sRed[5] + sRed[6] + sRed[7]);
    }
}

__global__ void msssim_combine(const float* __restrict__ accum,
                               float* __restrict__ out) {
    if (threadIdx.x == 0 && blockIdx.x == 0) {
        const double wts[5] = {0.0448, 0.2856, 0.3001, 0.2363, 0.1333};
        double res = 1.0;
        int H = 512;
        for (int l = 0; l < 5; ++l) {
            double cnt = 96.0 * (double)(H - 10) * (double)(H - 10);
            double ss = (double)accum[2 * l]     / cnt;
            double cs = (double)accum[2 * l + 1] / cnt;
            double v  = (l < 4) ? cs : ss;
            res *= pow(v, wts[l]);
            H >>= 1;
        }
        out[0] = (float)res;
    }
}

extern "C" void kernel_launch(void* const* d_in, const int* in_sizes, int n_in,
                              void* d_out, int out_size, void* d_ws, size_t ws_size,
                              hipStream_t stream) {
    const float* img1 = (const float*)d_in[0];
    const float* img2 = (const float*)d_in[1];
    float* ws    = (float*)d_ws;
    float* accum = ws;                      // 16 floats (10 used)

    const int Hs[5] = {512, 256, 128, 64, 32};
    float* p1[4]; float* p2[4];
    size_t off = 16;
    for (int l = 1; l <= 4; ++l) {
        size_t n = 96ull * (size_t)Hs[l] * (size_t)Hs[l];
        p1[l - 1] = ws + off; off += n;
        p2[l - 1] = ws + off; off += n;
    }

    msssim_zero_accum<<<dim3(1), dim3(32), 0, stream>>>(accum);

    for (int l = 0; l < 5; ++l) {
        int H = Hs[l];
        dim3 grid(H / TILE, H / TILE, 96);
        const float* a = (l == 0) ? img1 : p1[l - 1];
        const float* b = (l == 0) ? img2 : p2[l - 1];
        float* o1 = (l < 4) ? p1[l] : nullptr;
        float* o2 = (l < 4) ? p2[l] : nullptr;
        msssim_level<<<grid, dim3(128), 0, stream>>>(a, b, o1, o2,
                                                     accum + 2 * l, H, H);
    }

    msssim_combine<<<dim3(1), dim3(32), 0, stream>>>(accum, (float*)d_out);
}